// RPNLoss_75067438399749
// MI455X (gfx1250) — compile-verified
//
#include <hip/hip_runtime.h>

// ---------------------------------------------------------------------------
// RPN loss for MI455X (gfx1250, wave32).
//   R = 131072 regions, G = 64 ground truths.
//   Pass 1: TDM (tensor_load_to_lds) pulls the gt tile into LDS; per-pair
//           IoU + smooth-L1; threshold partials; per-gt argmax keys
//           (ds_max_u64 -> global_atomic_max_u64).
//   Pass 2: argmax corrections + final scalar loss.
// WMMA f32 16x16x4 (A = ones) performs the 32->16 lane fold of the block
// reduction on the matrix pipe; 4 shfl_xor steps finish the wave sum.
// ---------------------------------------------------------------------------

#define R_TOTAL   131072
#define G_TOTAL   64
#define T_PER_B   256
#define N_WAVES   (T_PER_B / 32)
#define IOU_POS_T 0.8f
#define IOU_NEG_T 0.3f
#define LAMBDA_C  10.0f
#define C1_LOG1P  0.3132616875182229f   // log1p(exp(-1))
#define LN2_C     0.6931471805599453f

typedef __attribute__((ext_vector_type(2))) float        v2f;
typedef __attribute__((ext_vector_type(8))) float        v8f;
typedef __attribute__((ext_vector_type(4))) unsigned int u32x4;
typedef __attribute__((ext_vector_type(8))) int          i32x8;
typedef __attribute__((ext_vector_type(4))) int          i32x4;

// Fold 32 lane values to 16 partial sums on the matrix pipe:
// D = A(=ones 16x4) x B + 0  =>  D[m][n] = sum_k B[k][n].
// With B.vgpr0 = v, B.vgpr1 = 0, lane L of d[0] holds v[L%16] + v[(L%16)+16].
__device__ __forceinline__ float wave_fold32to16(float v) {
    v2f a; a.x = 1.0f; a.y = 1.0f;
    v2f b; b.x = v;    b.y = 0.0f;
    v8f c = {};
    v8f d = __builtin_amdgcn_wmma_f32_16x16x4_f32(
        false, a, false, b, (short)0, c, false, false);
    return d[0];
}

// Full 32-lane sum: WMMA fold (32->16), then 4 xor-shuffle adds over the
// 16 distinct column sums. Every lane ends up with the wave total.
__device__ __forceinline__ float wave_sum(float v) {
    float s = wave_fold32to16(v);
    s += __shfl_xor(s, 1, 32);
    s += __shfl_xor(s, 2, 32);
    s += __shfl_xor(s, 4, 32);
    s += __shfl_xor(s, 8, 32);
    return s;
}

__device__ __forceinline__ float smooth_l1_coord(float d) {
    d = fabsf(d);
    return (d < 1.0f) ? 0.5f * d * d : d - 0.5f;
}

__global__ __launch_bounds__(T_PER_B)
void rpn_pass1(const float* __restrict__ scores,
               const float* __restrict__ regions,
               const float* __restrict__ anchors,
               const float* __restrict__ gts,
               float* __restrict__ g_acc,                 // [5] global accumulators
               unsigned long long* __restrict__ g_keys)   // [G] argmax keys
{
    __shared__ float4 s_gt_raw[G_TOTAL];                  // TDM destination (1 KB)
    __shared__ float gx1[G_TOTAL], gy1[G_TOTAL], gx2[G_TOTAL], gy2[G_TOTAL];
    __shared__ float gcx[G_TOTAL], gcy[G_TOTAL], glw[G_TOTAL], glh[G_TOTAL];
    __shared__ float garea[G_TOTAL];
    __shared__ unsigned long long s_key[G_TOTAL];
    __shared__ float s_wred[N_WAVES][5];

    const int tid  = threadIdx.x;
    const int lane = tid & 31;
    const int wid  = tid >> 5;

    // --- TDM: DMA the 64x4 f32 gt tile into LDS (wave 0 only; TDM ignores
    // EXEC, so the guard must be at wave granularity). D# per ISA 8.3/8.4:
    // 1D-style 256-element f32 tensor, tile 256x1.
    if (wid == 0) {
        const unsigned long long ga = (unsigned long long)(uintptr_t)gts;
        const unsigned lds_off = (unsigned)(uintptr_t)&s_gt_raw[0];
        u32x4 d0;
        d0.x = 1u;                                         // count=1 (valid), user mode
        d0.y = lds_off;                                    // lds_addr[31:0]
        d0.z = (unsigned)(ga & 0xffffffffull);             // global_addr[31:0]
        d0.w = (unsigned)((ga >> 32) & 0x01ffffffull)      // global_addr[56:32]
             | 0x80000000u;                                // type = 2 ("image")
        i32x8 d1;
        d1[0] = 0x00020000;          // workgroup_mask=0, data_size=2 (4 bytes)
        d1[1] = 0x01000000;          // tensor_dim0[15:0]=256 (bits 63:48)
        d1[2] = 0x00010000;          // tensor_dim1[15:0]=1   (bits 95:80)
        d1[3] = 0x01000000;          // tile_dim0=256          (bits 127:112)
        d1[4] = 0x00000001;          // tile_dim1=1, tile_dim2=0
        d1[5] = 0x00000100;          // tensor_dim0_stride=256 (low 32)
        d1[6] = 0x01000000;          // dim0_stride hi=0, tensor_dim1_stride lo=256
        d1[7] = 0x00000000;
        i32x4 z4 = {0, 0, 0, 0};                 // groups 2/3: dims 3/4 unused
        i32x8 z8 = {0, 0, 0, 0, 0, 0, 0, 0};     // extra group (clang-23 form)
        __builtin_amdgcn_tensor_load_to_lds(d0, d1, z4, z4, z8, 0);
        __builtin_amdgcn_s_wait_tensorcnt(0);
    }
    if (tid < G_TOTAL) s_key[tid] = 0ull;
    __syncthreads();

    // Derive per-gt quantities from the TDM-loaded tile.
    if (tid < G_TOTAL) {
        const float4 gb = s_gt_raw[tid];
        gx1[tid] = gb.x; gy1[tid] = gb.y; gx2[tid] = gb.z; gy2[tid] = gb.w;
        gcx[tid] = 0.5f * (gb.x + gb.z);
        gcy[tid] = 0.5f * (gb.y + gb.w);
        glw[tid] = logf(gb.z - gb.x);
        glh[tid] = logf(gb.w - gb.y);
        garea[tid] = (gb.z - gb.x) * (gb.w - gb.y);
    }
    __syncthreads();

    const int r = blockIdx.x * T_PER_B + tid;
    const float4 rb = ((const float4*)regions)[r];
    const float4 ab = ((const float4*)anchors)[r];

    const float area_r = (rb.z - rb.x) * (rb.w - rb.y);
    const float acx = 0.5f * (ab.x + ab.z);
    const float acy = 0.5f * (ab.y + ab.w);
    const float inv_aw = 1.0f / (ab.z - ab.x);
    const float inv_ah = 1.0f / (ab.w - ab.y);
    const float law = logf(ab.z - ab.x);
    const float lah = logf(ab.w - ab.y);

    // region parameterization (tx, ty, tw, th)
    const float rp0 = (0.5f * (rb.x + rb.z) - acx) * inv_aw;
    const float rp1 = (0.5f * (rb.y + rb.w) - acy) * inv_ah;
    const float rp2 = logf((rb.z - rb.x) * inv_aw);
    const float rp3 = logf((rb.w - rb.y) * inv_ah);

    float max_iou = 0.0f;
    float sl1_sum = 0.0f;
    float pair_cnt = 0.0f;

    #pragma unroll 4
    for (int g = 0; g < G_TOTAL; ++g) {
        // IoU
        const float ix1 = fmaxf(rb.x, gx1[g]);
        const float iy1 = fmaxf(rb.y, gy1[g]);
        const float ix2 = fminf(rb.z, gx2[g]);
        const float iy2 = fminf(rb.w, gy2[g]);
        const float inter = fmaxf(ix2 - ix1, 0.0f) * fmaxf(iy2 - iy1, 0.0f);
        const float iou = inter / (area_r + garea[g] - inter);
        max_iou = fmaxf(max_iou, iou);

        // gt parameterization against this thread's anchor
        const float t0 = (gcx[g] - acx) * inv_aw;
        const float t1 = (gcy[g] - acy) * inv_ah;
        const float t2 = glw[g] - law;
        const float t3 = glh[g] - lah;
        const float sl1 = 0.25f * (smooth_l1_coord(rp0 - t0) +
                                   smooth_l1_coord(rp1 - t1) +
                                   smooth_l1_coord(rp2 - t2) +
                                   smooth_l1_coord(rp3 - t3));
        if (iou > IOU_POS_T) { sl1_sum += sl1; pair_cnt += 1.0f; }

        // per-gt argmax over the wave: value-only max fold, ties push keys.
        float wmax = iou;
        #pragma unroll
        for (int m = 16; m >= 1; m >>= 1)
            wmax = fmaxf(wmax, __shfl_xor(wmax, m, 32));
        if (iou == wmax) {
            // monotonic key: larger iou wins; on ties larger ~r (smaller r) wins.
            const unsigned long long key =
                ((unsigned long long)__float_as_uint(iou) << 32) |
                (unsigned long long)(unsigned int)(~r);
            atomicMax(&s_key[g], key);
        }
    }

    // threshold-based row labels (argmax corrections applied in pass 2)
    const float score = scores[r];
    const bool has_pos = (max_iou > IOU_POS_T);
    const bool all_neg = (max_iou < IOU_NEG_T);
    const float cls_v = has_pos ? (1.0f - score + C1_LOG1P) : 0.0f;
    const float pos_v = has_pos ? 1.0f : 0.0f;
    const float neg_v = all_neg ? 1.0f : 0.0f;

    // Block reduction: matrix-pipe fold + shuffles; lane 0 stores per-wave
    // totals (plain ds_store -- avoids the serialized atomic-optimizer path).
    const float w0 = wave_sum(sl1_sum);
    const float w1 = wave_sum(pair_cnt);
    const float w2 = wave_sum(cls_v);
    const float w3 = wave_sum(pos_v);
    const float w4 = wave_sum(neg_v);
    if (lane == 0) {
        s_wred[wid][0] = w0;
        s_wred[wid][1] = w1;
        s_wred[wid][2] = w2;
        s_wred[wid][3] = w3;
        s_wred[wid][4] = w4;
    }
    __syncthreads();

    if (tid < 5) {                      // divergent addresses: no readlane loop
        float a = 0.0f;
        #pragma unroll
        for (int w = 0; w < N_WAVES; ++w) a += s_wred[w][tid];
        unsafeAtomicAdd(&g_acc[tid], a);
    }
    if (tid < G_TOTAL) atomicMax(&g_keys[tid], s_key[tid]);
}

__global__ __launch_bounds__(G_TOTAL)
void rpn_pass2(const float* __restrict__ scores,
               const float* __restrict__ regions,
               const float* __restrict__ anchors,
               const float* __restrict__ gts,
               const float* __restrict__ g_acc,
               const unsigned long long* __restrict__ g_keys,
               float* __restrict__ out)
{
    __shared__ float gx1[G_TOTAL], gy1[G_TOTAL], gx2[G_TOTAL], gy2[G_TOTAL];
    __shared__ float gcx[G_TOTAL], gcy[G_TOTAL], glw[G_TOTAL], glh[G_TOTAL];
    __shared__ float garea[G_TOTAL];
    __shared__ int   bestR[G_TOTAL];
    __shared__ float bestV[G_TOTAL];
    __shared__ float extra[5];   // [0]=reg, [1]=cnt, [2]=pos, [3]=neg, [4]=cls

    const int g = threadIdx.x;
    {
        const float4 gb = ((const float4*)gts)[g];
        gx1[g] = gb.x; gy1[g] = gb.y; gx2[g] = gb.z; gy2[g] = gb.w;
        gcx[g] = 0.5f * (gb.x + gb.z);
        gcy[g] = 0.5f * (gb.y + gb.w);
        glw[g] = logf(gb.z - gb.x);
        glh[g] = logf(gb.w - gb.y);
        garea[g] = (gb.z - gb.x) * (gb.w - gb.y);
        const unsigned long long k = g_keys[g];
        bestR[g] = (int)(~(unsigned int)(k & 0xffffffffull));
        bestV[g] = __uint_as_float((unsigned int)(k >> 32));
    }
    if (g < 5) extra[g] = 0.0f;
    __syncthreads();

    const int r = bestR[g];
    const float4 rb = ((const float4*)regions)[r];
    const float4 ab = ((const float4*)anchors)[r];
    const float area_r = (rb.z - rb.x) * (rb.w - rb.y);
    const float acx = 0.5f * (ab.x + ab.z);
    const float acy = 0.5f * (ab.y + ab.w);
    const float inv_aw = 1.0f / (ab.z - ab.x);
    const float inv_ah = 1.0f / (ab.w - ab.y);
    const float law = logf(ab.z - ab.x);
    const float lah = logf(ab.w - ab.y);

    // Pair correction: (best[g], g) is forced to label 1; add its smooth-L1 if
    // it wasn't already above the positive IoU threshold in pass 1.
    if (bestV[g] <= IOU_POS_T) {
        const float rp0 = (0.5f * (rb.x + rb.z) - acx) * inv_aw;
        const float rp1 = (0.5f * (rb.y + rb.w) - acy) * inv_ah;
        const float rp2 = logf((rb.z - rb.x) * inv_aw);
        const float rp3 = logf((rb.w - rb.y) * inv_ah);
        const float t0 = (gcx[g] - acx) * inv_aw;
        const float t1 = (gcy[g] - acy) * inv_ah;
        const float t2 = glw[g] - law;
        const float t3 = glh[g] - lah;
        const float sl1 = 0.25f * (smooth_l1_coord(rp0 - t0) +
                                   smooth_l1_coord(rp1 - t1) +
                                   smooth_l1_coord(rp2 - t2) +
                                   smooth_l1_coord(rp3 - t3));
        atomicAdd(&extra[0], sl1);
        atomicAdd(&extra[1], 1.0f);
    }

    // Row correction (dedup repeated best regions): promote to pos / undo neg.
    bool first = true;
    for (int j = 0; j < g; ++j) if (bestR[j] == r) { first = false; break; }
    if (first) {
        float max_iou = 0.0f;
        for (int j = 0; j < G_TOTAL; ++j) {
            const float ix1 = fmaxf(rb.x, gx1[j]);
            const float iy1 = fmaxf(rb.y, gy1[j]);
            const float ix2 = fminf(rb.z, gx2[j]);
            const float iy2 = fminf(rb.w, gy2[j]);
            const float inter = fmaxf(ix2 - ix1, 0.0f) * fmaxf(iy2 - iy1, 0.0f);
            max_iou = fmaxf(max_iou, inter / (area_r + garea[j] - inter));
        }
        if (max_iou <= IOU_POS_T) {          // was not pos in pass 1; now pos
            atomicAdd(&extra[2], 1.0f);
            atomicAdd(&extra[4], 1.0f - scores[r] + C1_LOG1P);
        }
        if (max_iou < IOU_NEG_T)             // was counted neg in pass 1; undo
            atomicAdd(&extra[3], -1.0f);
    }
    __syncthreads();

    if (g == 0) {
        const float reg_sum = g_acc[0] + extra[0];
        const float reg_cnt = g_acc[1] + extra[1];
        const float cls_sum = g_acc[2] + extra[4];
        const float n_pos   = g_acc[3] + extra[2];
        const float n_neg   = g_acc[4] + extra[3];
        const float n = n_pos + n_neg;
        const float cls_loss = (cls_sum + n_neg * LN2_C) / n;
        out[0] = cls_loss / n + LAMBDA_C * reg_sum / reg_cnt;  // K = 1
        out[1] = (float)bestR[G_TOTAL - 1];                    // max_iou_idx
    }
}

__global__ void rpn_init(float* __restrict__ g_acc,
                         unsigned long long* __restrict__ g_keys)
{
    const int t = threadIdx.x;
    if (t < 8) g_acc[t] = 0.0f;
    if (t < G_TOTAL) g_keys[t] = 0ull;
}

extern "C" void kernel_launch(void* const* d_in, const int* in_sizes, int n_in,
                              void* d_out, int out_size, void* d_ws, size_t ws_size,
                              hipStream_t stream)
{
    (void)in_sizes; (void)n_in; (void)out_size; (void)ws_size;
    const float* scores  = (const float*)d_in[0];
    const float* regions = (const float*)d_in[1];
    const float* anchors = (const float*)d_in[2];
    const float* gts     = (const float*)d_in[3];
    float* out = (float*)d_out;

    float* g_acc = (float*)d_ws;                                          // 8 floats
    unsigned long long* g_keys = (unsigned long long*)((char*)d_ws + 64); // 64 u64

    rpn_init<<<1, 64, 0, stream>>>(g_acc, g_keys);
    rpn_pass1<<<R_TOTAL / T_PER_B, T_PER_B, 0, stream>>>(
        scores, regions, anchors, gts, g_acc, g_keys);
    rpn_pass2<<<1, G_TOTAL, 0, stream>>>(
        scores, regions, anchors, gts, g_acc, g_keys, out);
}